// EyeLoss_83691732730572
// MI455X (gfx1250) — compile-verified
//
#include <hip/hip_runtime.h>

// CDNA5 (gfx1250) WMMA operand types
typedef __attribute__((ext_vector_type(2))) float v2f;   // A/B for 16x16x4 f32: 64 f32 / 32 lanes
typedef __attribute__((ext_vector_type(8))) float v8f;   // 16x16 f32 C/D accumulator

#define HM_L   98
#define HM_HW  4096   // 64*64
#define HM_W   64

__device__ __forceinline__ float sgnf(float x) {
    return (x > 0.0f) ? 1.0f : ((x < 0.0f) ? -1.0f : 0.0f);
}

// Order-preserving f32 -> u32 map (no NaNs in input), packed with ~idx so that
// u64-max == (max value, lowest index) == jnp.argmax first-occurrence rule.
__device__ __forceinline__ unsigned long long packkey(float f, int idx) {
    unsigned u = __float_as_uint(f);
    unsigned m = (u & 0x80000000u) ? ~u : (u | 0x80000000u);
    return ((unsigned long long)m << 32) | (unsigned)(~idx);
}

// One block per (batch, channel) pair, channel in {96, 97}. 256 threads.
// argmax(src), argmax(tgt) via packed-key ds_max_u64; sum((src-tgt)^2) via
// V_WMMA_F32_16X16X4_F32 row-sum trick (B = all-ones).
__global__ void __launch_bounds__(256)
eye_block_kernel(const float* __restrict__ src,
                 const float* __restrict__ tgt,
                 float* __restrict__ out,       // [0]=loss (kernel2), [1..512]=src eyes, [513..1024]=tgt eyes
                 float* __restrict__ partial) { // 256 per-block squared-diff sums
    const int tid = threadIdx.x;        // 0..255
    const int bc  = blockIdx.x;         // 0..255
    const int b   = bc >> 1;
    const int c   = 96 + (bc & 1);
    const long base = (long)(b * HM_L + c) * HM_HW;

    const float* __restrict__ sp = src + base;
    const float* __restrict__ tp = tgt + base;

    // gfx1250 prefetch path (global_prefetch_b8)
    __builtin_prefetch(sp + tid * 16, 0, 0);
    __builtin_prefetch(tp + tid * 16, 0, 0);

    __shared__ unsigned long long smax, tmax;   // packed argmax keys
    __shared__ float wl[16];                    // per-wave WMMA row sums (lanes 0 & 16 of 8 waves)

    if (tid == 0) { smax = 0ull; tmax = 0ull; } // 0 < packkey(x, i) for all finite x
    __syncthreads();

    unsigned long long skey = 0ull, tkey = 0ull;

    v8f acc = {};                                // WMMA C/D accumulator (f32)
    v2f ones; ones[0] = 1.0f; ones[1] = 1.0f;    // B = all-ones -> D rows = row sums of A

    #pragma unroll
    for (int i = 0; i < 4; ++i) {
        const int off = i * 1024 + tid * 4;      // fully coalesced float4 sweeps
        const float4 s4 = *reinterpret_cast<const float4*>(sp + off);
        const float4 t4 = *reinterpret_cast<const float4*>(tp + off);

        // branchless packed-key argmax tracking
        unsigned long long k;
        k = packkey(s4.x, off + 0); skey = (k > skey) ? k : skey;
        k = packkey(s4.y, off + 1); skey = (k > skey) ? k : skey;
        k = packkey(s4.z, off + 2); skey = (k > skey) ? k : skey;
        k = packkey(s4.w, off + 3); skey = (k > skey) ? k : skey;
        k = packkey(t4.x, off + 0); tkey = (k > tkey) ? k : tkey;
        k = packkey(t4.y, off + 1); tkey = (k > tkey) ? k : tkey;
        k = packkey(t4.z, off + 2); tkey = (k > tkey) ? k : tkey;
        k = packkey(t4.w, off + 3); tkey = (k > tkey) ? k : tkey;

        const float d0 = s4.x - t4.x, d1 = s4.y - t4.y;
        const float d2 = s4.z - t4.z, d3 = s4.w - t4.w;
        v2f a; a[0] = d0 * d0 + d1 * d1;
               a[1] = d2 * d2 + d3 * d3;
        // acc = a x ones + acc  (v_wmma_f32_16x16x4_f32); EXEC is all-1s here.
        acc = __builtin_amdgcn_wmma_f32_16x16x4_f32(
            /*neg_a=*/false, a, /*neg_b=*/false, ones,
            /*c_mod=*/(short)0, acc, /*reuse_a=*/false, /*reuse_b=*/false);
    }

    // Cross-thread argmax: order-independent LDS max atomics (ds_max_u64).
    atomicMax(&smax, skey);
    atomicMax(&tmax, tkey);

    // C/D 16x16 layout: column n replicated; lane 0 holds rows 0..7, lane 16 rows 8..15.
    // Block partial sum = sum over waves of rowsum(lane0) + rowsum(lane16).
    const float rs = acc[0] + acc[1] + acc[2] + acc[3] + acc[4] + acc[5] + acc[6] + acc[7];
    const int lane = tid & 31, w = tid >> 5;
    if (lane == 0)  wl[w * 2 + 0] = rs;
    if (lane == 16) wl[w * 2 + 1] = rs;
    __syncthreads();

    if (tid == 0) {
        // Deterministic fixed-order combine of the 16 WMMA partial sums.
        float bsum = 0.0f;
        #pragma unroll
        for (int i = 0; i < 16; ++i) bsum += wl[i];
        partial[bc] = bsum;

        const int sidx = ~(int)(unsigned)smax & 0xFFF;   // low 32 bits = ~idx, idx in [0,4095]
        const int tidx = ~(int)(unsigned)tmax & 0xFFF;

        // Sub-pixel refinement + eyes output, matching the reference exactly.
        auto refine = [&](const float* __restrict__ p, int idx, float& ox, float& oy) {
            const int px = idx & (HM_W - 1);
            const int py = idx >> 6;
            auto g = [&](int dx, int dy) {
                const int cx = min(max(px + dx, 0), HM_W - 1);
                const int cy = min(max(py + dy, 0), HM_W - 1);
                return p[cy * HM_W + cx];
            };
            const float dxv = g(1, 0) - g(-1, 0);
            const float dyv = g(0, 1) - g(0, -1);
            const bool inside = (px > 0) && (px < HM_W - 1) && (py > 0) && (py < HM_W - 1);
            const float offx = inside ? sgnf(dxv) * 0.25f : 0.0f;
            const float offy = inside ? sgnf(dyv) * 0.25f : 0.0f;
            // preds = p0 + 1 + off - 0.5; landmarks = preds * 4
            ox = ((float)px + 0.5f + offx) * 4.0f;
            oy = ((float)py + 0.5f + offy) * 4.0f;
        };

        float sx, sy, tx, ty;
        refine(sp, sidx, sx, sy);
        refine(tp, tidx, tx, ty);
        const int eo = b * 4 + (c - 96) * 2;     // eyes = [x96,y96,x97,y97] per batch
        out[1 + eo]       = sx;
        out[1 + eo + 1]   = sy;
        out[513 + eo]     = tx;
        out[513 + eo + 1] = ty;
    }
}

// Deterministic final reduction of the 256 per-block partial sums.
__global__ void __launch_bounds__(256)
loss_reduce_kernel(const float* __restrict__ partial, float* __restrict__ out) {
    __shared__ float sm[256];
    const int tid = threadIdx.x;
    sm[tid] = partial[tid];
    __syncthreads();
    for (int s = 128; s > 0; s >>= 1) {
        if (tid < s) sm[tid] += sm[tid + s];
        __syncthreads();
    }
    if (tid == 0) {
        // left + right = (sum96 + sum97) / (B*H*W) with B*H*W = 128*64*64 = 524288
        out[0] = sm[0] * (1.0f / 524288.0f);
    }
}

extern "C" void kernel_launch(void* const* d_in, const int* in_sizes, int n_in,
                              void* d_out, int out_size, void* d_ws, size_t ws_size,
                              hipStream_t stream) {
    const float* src = (const float*)d_in[0];   // source_heatmap [128,98,64,64] f32
    const float* tgt = (const float*)d_in[1];   // target_heatmap [128,98,64,64] f32
    float* out     = (float*)d_out;             // 1 + 512 + 512 = 1025 f32
    float* partial = (float*)d_ws;              // 256 f32 scratch

    eye_block_kernel<<<256, 256, 0, stream>>>(src, tgt, out, partial);
    loss_reduce_kernel<<<1, 256, 0, stream>>>(partial, out);
}